// MultiAttention_3006477107709
// MI455X (gfx1250) — compile-verified
//
#include <hip/hip_runtime.h>

typedef __attribute__((ext_vector_type(2))) float v2f;
typedef __attribute__((ext_vector_type(8))) float v8f;

#define CONTEXT 5
#define DIM 32
#define HEADS 4
#define HEAD_DIM 8
#define ROWS 16   // padded M dimension for 16x16 WMMA tiles

// D-tile accumulate: D(16x16) += A(16x4) * B(4x16), 8 K-steps over K=32.
// A comes from LDS (row-major [16][32]); B is W^T on the fly from the global
// weight matrix W[32][32] (nn.Linear: y = x @ W^T, so B[k][n] = W[n][k]).
__device__ __forceinline__ v8f mm_tile_xwt(const float* __restrict__ Asrc,
                                           const float* __restrict__ W,
                                           int ntile, int lane) {
  const int lo = lane & 15;
  const int hi = lane >> 4;
  v8f c = {};
#pragma unroll
  for (int kk = 0; kk < 8; ++kk) {
    const int k0 = 4 * kk + 2 * hi;  // per-lane K base per ISA A/B f32 layout
    v2f a, b;
    a.x = Asrc[lo * DIM + k0 + 0];
    a.y = Asrc[lo * DIM + k0 + 1];
    b.x = W[(ntile * 16 + lo) * DIM + k0 + 0];
    b.y = W[(ntile * 16 + lo) * DIM + k0 + 1];
    // 8 args: (neg_a, A, neg_b, B, c_mod, C, reuse_a, reuse_b)
    c = __builtin_amdgcn_wmma_f32_16x16x4_f32(false, a, false, b,
                                              (short)0, c, false, false);
  }
  return c;
}

// Scatter a 16x16 f32 D tile into a row-major LDS [16][32] buffer.
__device__ __forceinline__ void store_tile(float* __restrict__ Dst, v8f c,
                                           int ntile, int lane) {
  const int lo = lane & 15;
  const int hi = lane >> 4;
#pragma unroll
  for (int v = 0; v < 8; ++v)
    Dst[(v + 8 * hi) * DIM + ntile * 16 + lo] = c[v];
}

__global__ __launch_bounds__(64) void attn_block_kernel(
    const float* __restrict__ X, const float* __restrict__ WQ,
    const float* __restrict__ WK, const float* __restrict__ WV,
    const float* __restrict__ Wproj, const float* __restrict__ bproj,
    const float* __restrict__ W1, const float* __restrict__ b1,
    const float* __restrict__ W2, const float* __restrict__ b2,
    float* __restrict__ out) {
  __shared__ float Xs[ROWS * DIM];
  __shared__ float Qs[ROWS * DIM];
  __shared__ float Ks[ROWS * DIM];
  __shared__ float Vs[ROWS * DIM];
  __shared__ float Ys[ROWS * DIM];  // concatenated attention output
  __shared__ float Ps[ROWS * DIM];  // Yproj (needed again for residual)
  __shared__ float Hs[ROWS * DIM];  // relu MLP hidden

  const int tid = threadIdx.x;   // 0..63 (two wave32s)
  const int lane = tid & 31;
  const int ntile = tid >> 5;    // wave id selects N-tile (wave-uniform)

  // Stage X into LDS, zero the pad rows (and intermediates, for determinism).
  for (int i = tid; i < ROWS * DIM; i += 64) {
    Xs[i] = (i < CONTEXT * DIM) ? X[i] : 0.0f;
    Qs[i] = 0.0f; Ks[i] = 0.0f; Vs[i] = 0.0f;
    Ys[i] = 0.0f; Ps[i] = 0.0f; Hs[i] = 0.0f;
  }
  __syncthreads();

  // ---- QKV projections: Q = X @ WQ^T, etc. (fp32 WMMA) ----
  store_tile(Qs, mm_tile_xwt(Xs, WQ, ntile, lane), ntile, lane);
  store_tile(Ks, mm_tile_xwt(Xs, WK, ntile, lane), ntile, lane);
  store_tile(Vs, mm_tile_xwt(Xs, WV, ntile, lane), ntile, lane);
  __syncthreads();

  // ---- Causal multi-head attention, one (head, query) pair per lane ----
  if (tid < HEADS * CONTEXT) {
    const int h = tid / CONTEXT;
    const int q = tid % CONTEXT;
    const float scale = 0.17677669529663687f;  // 1/sqrt(DIM) per reference
    float s[CONTEXT];
    float mx = -3.4e38f;
    for (int k = 0; k <= q; ++k) {           // causal: k <= q only
      float acc = 0.0f;
      for (int d = 0; d < HEAD_DIM; ++d)
        acc += Qs[q * DIM + h * HEAD_DIM + d] * Ks[k * DIM + h * HEAD_DIM + d];
      acc *= scale;
      s[k] = acc;
      mx = fmaxf(mx, acc);
    }
    float sum = 0.0f;
    for (int k = 0; k <= q; ++k) {
      s[k] = expf(s[k] - mx);
      sum += s[k];
    }
    const float inv = 1.0f / sum;
    for (int d = 0; d < HEAD_DIM; ++d) {
      float y = 0.0f;
      for (int k = 0; k <= q; ++k)
        y += s[k] * Vs[k * DIM + h * HEAD_DIM + d];
      Ys[q * DIM + h * HEAD_DIM + d] = y * inv;
    }
  }
  __syncthreads();

  // ---- Output projection: Yproj = Ycat @ Wproj^T + bproj ----
  {
    v8f c = mm_tile_xwt(Ys, Wproj, ntile, lane);
    const float bias = bproj[ntile * 16 + (lane & 15)];  // same N for all 8 regs
#pragma unroll
    for (int v = 0; v < 8; ++v) c[v] += bias;
    store_tile(Ps, c, ntile, lane);
  }
  __syncthreads();

  // ---- MLP hidden: H1 = relu(Yproj @ W1^T + b1) ----
  {
    v8f c = mm_tile_xwt(Ps, W1, ntile, lane);
    const float bias = b1[ntile * 16 + (lane & 15)];
#pragma unroll
    for (int v = 0; v < 8; ++v) c[v] = fmaxf(c[v] + bias, 0.0f);
    store_tile(Hs, c, ntile, lane);
  }
  __syncthreads();

  // ---- Y2 = H1 @ W2^T + b2; out = Y2 + Yproj (residual) ----
  {
    v8f c = mm_tile_xwt(Hs, W2, ntile, lane);
    const int lo = lane & 15;
    const int hi = lane >> 4;
    const int n = ntile * 16 + lo;
    const float bias = b2[n];
#pragma unroll
    for (int v = 0; v < 8; ++v) {
      const int m = v + 8 * hi;
      if (m < CONTEXT)
        out[m * DIM + n] = c[v] + bias + Ps[m * DIM + n];
    }
  }
}

extern "C" void kernel_launch(void* const* d_in, const int* in_sizes, int n_in,
                              void* d_out, int out_size, void* d_ws, size_t ws_size,
                              hipStream_t stream) {
  (void)in_sizes; (void)n_in; (void)out_size; (void)d_ws; (void)ws_size;
  const float* X     = (const float*)d_in[0];
  const float* WQ    = (const float*)d_in[1];
  const float* WK    = (const float*)d_in[2];
  const float* WV    = (const float*)d_in[3];
  const float* Wproj = (const float*)d_in[4];
  const float* bproj = (const float*)d_in[5];
  const float* W1    = (const float*)d_in[6];
  const float* b1    = (const float*)d_in[7];
  const float* W2    = (const float*)d_in[8];
  const float* b2    = (const float*)d_in[9];
  float* out = (float*)d_out;

  attn_block_kernel<<<1, 64, 0, stream>>>(X, WQ, WK, WV, Wproj, bproj,
                                          W1, b1, W2, b2, out);
}